// NetSpacing_51634096832986
// MI455X (gfx1250) — compile-verified
//
#include <hip/hip_runtime.h>
#include <math.h>

// ---------------------------------------------------------------------------
// NetSpacing weighted-average wirelength + direction penalty, MI455X (gfx1250)
//
// Bandwidth-bound (~85 MB traffic, ~0.2 GFLOP -> ~4us floor at 23.3 TB/s).
// pin2net_map is sorted => nets own contiguous pin ranges => CSR + one thread
// per net, no atomics, fully deterministic reduction.
//
// CDNA5 specifics used:
//  - v_wmma_f32_16x16x4_f32 with all-ones B as a single-instruction wave32
//    cross-lane sum (row sums of A), + one shfl_xor(16) to fold halves.
//  - global_prefetch_b8 via __builtin_prefetch to warm pin_dir for pass 2.
//  - wave32 layout assumptions (8 waves / 256-thread block), LDS block reduce.
// ---------------------------------------------------------------------------

#define TB        256
#define GAMMA_INV 0.25f      // 1/GAMMA, GAMMA = 4.0
#define CTHRESH   0.5f
#define EPS_WA    1e-12f

typedef __attribute__((ext_vector_type(2))) float v2f;
typedef __attribute__((ext_vector_type(8))) float v8f;

// Wave32 sum via WMMA: A holds one value per lane (second slot 0), B = ones.
// D[m][n] = rowsum_m for every n; lanes 0..15 VGPRs 0..7 hold rows 0..7,
// lanes 16..31 hold rows 8..15. Summing a lane's 8 D registers gives half the
// wave total; shfl_xor(16) folds in the other half. Result valid in all lanes.
__device__ __forceinline__ float wave_reduce_sum_wmma(float v) {
    v2f a; a[0] = v;    a[1] = 0.0f;
    v2f b; b[0] = 1.0f; b[1] = 1.0f;
    v8f c = {};
    v8f d = __builtin_amdgcn_wmma_f32_16x16x4_f32(
        /*neg_a=*/false, a, /*neg_b=*/false, b,
        /*c_mod=*/(short)0, c, /*reuse_a=*/false, /*reuse_b=*/false);
    float r = d[0] + d[1] + d[2] + d[3] + d[4] + d[5] + d[6] + d[7];
    r += __shfl_xor(r, 16, 32);
    return r;
}

// Deterministic 256-thread block reduction (8 wave32 waves). Result in tid 0.
__device__ __forceinline__ float block_reduce_sum(float v) {
    __shared__ float lds[TB / 32];
    float w = wave_reduce_sum_wmma(v);          // EXEC all-1s here by construction
    const int lane = threadIdx.x & 31;
    const int wid  = threadIdx.x >> 5;
    if (lane == 0) lds[wid] = w;
    __syncthreads();
    float s = 0.0f;
    if (threadIdx.x == 0) {
        #pragma unroll
        for (int i = 0; i < TB / 32; ++i) s += lds[i];   // fixed order
    }
    return s;
}

// --- Kernel 0: zero CSR arrays (workspace is poisoned; must re-init per call)
__global__ void ns_init_csr(int* __restrict__ starts, int* __restrict__ ends, int N) {
    int n = blockIdx.x * blockDim.x + threadIdx.x;
    if (n < N) { starts[n] = 0; ends[n] = 0; }
}

// --- Kernel 1: boundary detect on the sorted pin2net map -> CSR [start,end)
__global__ void ns_build_csr(const int* __restrict__ seg,
                             int* __restrict__ starts, int* __restrict__ ends, int P) {
    int i = blockIdx.x * blockDim.x + threadIdx.x;
    if (i >= P) return;
    int s = seg[i];
    if (i == 0     || seg[i - 1] != s) starts[s] = i;
    if (i == P - 1 || seg[i + 1] != s) ends[s]   = i + 1;
}

// --- Kernel 2: one thread per net; two passes over its contiguous pin range.
// Pass-2 re-reads of x/y hit L0 (a wave covers ~1KB contiguous), so HBM sees
// each byte once. Per-block partial is reduced deterministically (WMMA + LDS).
__global__ void ns_per_net(const float* __restrict__ x,  const float* __restrict__ y,
                           const float* __restrict__ pdx, const float* __restrict__ pdy,
                           const float* __restrict__ w,   const unsigned char* __restrict__ mask,
                           const int* __restrict__ starts, const int* __restrict__ ends,
                           float* __restrict__ partials, int N) {
    int n = blockIdx.x * blockDim.x + threadIdx.x;
    float val = 0.0f;
    if (n < N) {
        const int s = starts[n];
        const int e = ends[n];
        if (e > s) {
            // Warm pin_dir for pass 2 (global_prefetch_b8).
            __builtin_prefetch(&pdx[s], 0, 1);
            __builtin_prefetch(&pdy[s], 0, 1);

            // Pass 1: segment max/min + sums.
            float maxx = -INFINITY, minx = INFINITY;
            float maxy = -INFINITY, miny = INFINITY;
            float sx = 0.0f, sy = 0.0f;
            for (int i = s; i < e; ++i) {
                float xv = x[i], yv = y[i];
                maxx = fmaxf(maxx, xv); minx = fminf(minx, xv);
                maxy = fmaxf(maxy, yv); miny = fminf(miny, yv);
                sx += xv; sy += yv;
            }
            const float cnt     = (float)(e - s);
            const float inv_cnt = 1.0f / cnt;
            const float cx = sx * inv_cnt, cy = sy * inv_cnt;

            // Pass 2: stabilized exp sums + direction penalty.
            float sex = 0.f, smx = 0.f, svex = 0.f, svmx = 0.f;
            float sey = 0.f, smy = 0.f, svey = 0.f, svmy = 0.f;
            float pen = 0.f;
            for (int i = s; i < e; ++i) {
                float xv = x[i], yv = y[i];
                float epx = __expf((xv - maxx) * GAMMA_INV);
                float emx = __expf((minx - xv) * GAMMA_INV);
                float epy = __expf((yv - maxy) * GAMMA_INV);
                float emy = __expf((miny - yv) * GAMMA_INV);
                sex += epx; svex += xv * epx;
                smx += emx; svmx += xv * emx;
                sey += epy; svey += yv * epy;
                smy += emy; svmy += yv * emy;
                float dxp = cx - xv, dyp = cy - yv;
                float dn  = sqrtf(dxp * dxp + dyp * dyp) + 1e-8f;
                float cs  = (dxp * pdx[i] + dyp * pdy[i]) / dn;
                pen += fmaxf(0.0f, CTHRESH - cs);
            }
            float wax = svex / fmaxf(sex, EPS_WA) - svmx / fmaxf(smx, EPS_WA);
            float way = svey / fmaxf(sey, EPS_WA) - svmy / fmaxf(smy, EPS_WA);
            float wl  = fmaxf(wax + way, 0.0f);              // ALPHA == 1
            val = w[n] * (mask[n] ? 1.0f : 0.0f) * (1.0f + pen * inv_cnt) * wl;
        }
    }
    // All threads (full EXEC) enter the reduction.
    float bs = block_reduce_sum(val);
    if (threadIdx.x == 0) partials[blockIdx.x] = bs;
}

// --- Kernel 3: single-block deterministic reduction of per-block partials.
__global__ void ns_final(const float* __restrict__ partials, int nPart,
                         float* __restrict__ out) {
    float acc = 0.0f;
    for (int i = threadIdx.x; i < nPart; i += blockDim.x) acc += partials[i]; // fixed order
    float s = block_reduce_sum(acc);
    if (threadIdx.x == 0) out[0] = s;
}

extern "C" void kernel_launch(void* const* d_in, const int* in_sizes, int n_in,
                              void* d_out, int out_size, void* d_ws, size_t ws_size,
                              hipStream_t stream) {
    // setup_inputs order: pos, pin_dir_x, pin_dir_y, net_weights, pin2net_map,
    //                     net_mask, pin_mask
    const float*         pos  = (const float*)d_in[0];
    const float*         pdx  = (const float*)d_in[1];
    const float*         pdy  = (const float*)d_in[2];
    const float*         w    = (const float*)d_in[3];
    const int*           seg  = (const int*)d_in[4];
    const unsigned char* mask = (const unsigned char*)d_in[5];
    const int P = in_sizes[1];   // NUM_PINS
    const int N = in_sizes[3];   // NUM_NETS

    const float* x = pos;
    const float* y = pos + P;

    // Workspace: starts[N] | ends[N] | partials[gridN]
    int*   starts   = (int*)d_ws;
    int*   ends     = starts + N;
    float* partials = (float*)(ends + N);

    const int gN = (N + TB - 1) / TB;
    const int gP = (P + TB - 1) / TB;

    ns_init_csr <<<gN, TB, 0, stream>>>(starts, ends, N);
    ns_build_csr<<<gP, TB, 0, stream>>>(seg, starts, ends, P);
    ns_per_net  <<<gN, TB, 0, stream>>>(x, y, pdx, pdy, w, mask,
                                        starts, ends, partials, N);
    ns_final    <<<1,  TB, 0, stream>>>(partials, gN, (float*)d_out);
}